// Rasterizer_32031866093902
// MI455X (gfx1250) — compile-verified
//
#include <hip/hip_runtime.h>
#include <limits.h>

#define HH 160
#define WW 160
#define BSZ 32
#define FCHUNK 256

#define INF_V   3.4028235e35f   /* FLT_MAX * 0.001  */
#define LOW_INF 3.4028235e34f   /* FLT_MAX * 0.0001 */
#define FTINY   1.17549435e-35f /* FLT_TINY * 1000  */

typedef __attribute__((ext_vector_type(2))) float v2f;
typedef __attribute__((ext_vector_type(4))) float v4f;
typedef __attribute__((ext_vector_type(8))) float v8f;
typedef __attribute__((ext_vector_type(4))) int   v4i;

// ---------------------------------------------------------------- init gbox
__global__ void rast_init_gbox(int* gbox, int B) {
    int t = threadIdx.x;
    if (t < B * 4) {
        int k = t & 3;
        gbox[t] = (k == 0 || k == 2) ? INT_MAX : INT_MIN;
    }
}

// -------------------------------------------------------- per-face setup
__global__ __launch_bounds__(256) void rast_setup(
    const float* __restrict__ pt2d, const float* __restrict__ color,
    const float* __restrict__ pt3d, const float* __restrict__ normal,
    const float* __restrict__ Rm,   const float* __restrict__ Tm,
    const int*   __restrict__ face,
    float* __restrict__ planes, float* __restrict__ colw,
    int* __restrict__ bbox, int* __restrict__ gbox,
    int Fnum, int FP, int Vnum)
{
    int gid = blockIdx.x * blockDim.x + threadIdx.x;
    int b = gid / FP;
    int f = gid - b * FP;

    float l0x=0,l0y=0,l0c=0,l1x=0,l1y=0,l1c=0,l2x=0,l2y=0,l2c=0;
    float Dx=0,Dy=0,Dc=0;
    float Cx[3]={0,0,0},Cy[3]={0,0,0},Cc[3]={0,0,0};
    bool valid = false;
    int pxmin=INT_MAX/2, pxmax=INT_MIN/2, pymin=INT_MAX/2, pymax=INT_MIN/2;

    if (f < Fnum) {
        int i0 = face[f], i1 = face[Fnum + f], i2 = face[2*Fnum + f];
        const float* p2 = pt2d + (size_t)b * 3 * Vnum;
        float x0 = p2[i0],        x1 = p2[i1],        x2 = p2[i2];
        float y0 = p2[Vnum+i0],   y1 = p2[Vnum+i1],   y2 = p2[Vnum+i2];
        float z0 = p2[2*Vnum+i0], z1 = p2[2*Vnum+i1], z2 = p2[2*Vnum+i2];

        const float* R = Rm + b * 9;
        const float* T = Tm + b * 3;
        float rt0 = R[0]*T[0] + R[3]*T[1] + R[6]*T[2];
        float rt1 = R[1]*T[0] + R[4]*T[1] + R[7]*T[2];
        float rt2 = R[2]*T[0] + R[5]*T[1] + R[8]*T[2];
        const float* p3 = pt3d + (size_t)b * 3 * Vnum;
        const float* nm = normal + (size_t)b * 3 * Fnum;
        float dn = (p3[i0]        + rt0) * nm[f]
                 + (p3[Vnum+i0]   + rt1) * nm[Fnum+f]
                 + (p3[2*Vnum+i0] + rt2) * nm[2*Fnum+f];
        valid = (dn < 0.0f) && (fminf(fminf(z0, z1), z2) > 0.0f);

        float det = (y1 - y2) * (x0 - x2) + (x2 - x1) * (y0 - y2);
        float sg  = (float)((det > 0.0f) - (det < 0.0f));
        det = sg * fmaxf(fabsf(det), FTINY);
        float inv = 1.0f / det;

        l0x = (y1 - y2) * inv;  l0y = (x2 - x1) * inv;
        l0c = -l0x * x2 - l0y * y2;
        l1x = (y2 - y0) * inv;  l1y = (x0 - x2) * inv;
        l1c = -l1x * x2 - l1y * y2;
        l2x = -l0x - l1x;       l2y = -l0y - l1y;
        l2c = 1.0f - l0c - l1c;

        Dx = z0*l0x + z1*l1x + z2*l2x;
        Dy = z0*l0y + z1*l1y + z2*l2y;
        Dc = z0*l0c + z1*l1c + z2*l2c;

        const float* cb = color + (size_t)b * 3 * Vnum;
        #pragma unroll
        for (int ch = 0; ch < 3; ++ch) {
            float c0 = cb[ch*Vnum + i0], c1 = cb[ch*Vnum + i1], c2 = cb[ch*Vnum + i2];
            Cx[ch] = c0*l0x + c1*l1x + c2*l2x;
            Cy[ch] = c0*l0y + c1*l1y + c2*l2y;
            Cc[ch] = c0*l0c + c1*l1c + c2*l2c;
        }

        if (valid) {
            int px0 = (int)x0, px1 = (int)x1, px2 = (int)x2;
            int py0 = (int)y0, py1 = (int)y1, py2 = (int)y2;
            pxmin = min(min(px0, px1), px2);  pxmax = max(max(px0, px1), px2);
            pymin = min(min(py0, py1), py2);  pymax = max(max(py0, py1), py2);
            atomicMin(&gbox[b*4+0], pxmin);  atomicMax(&gbox[b*4+1], pxmax);
            atomicMin(&gbox[b*4+2], pymin);  atomicMax(&gbox[b*4+3], pymax);
        }
    }

    float* P = planes + ((size_t)b * FP + f) * 16;
    P[0]=l0x; P[1]=l0y; P[2]=l0c; P[3]=0.0f;
    P[4]=l1x; P[5]=l1y; P[6]=l1c; P[7]=0.0f;
    P[8]=l2x; P[9]=l2y; P[10]=l2c; P[11]=0.0f;
    P[12]=Dx; P[13]=Dy; P[14]=Dc; P[15]=0.0f;

    float* C = colw + ((size_t)b * FP + f) * 12;
    #pragma unroll
    for (int ch = 0; ch < 3; ++ch) {
        C[ch*4+0]=Cx[ch]; C[ch*4+1]=Cy[ch]; C[ch*4+2]=Cc[ch]; C[ch*4+3]=0.0f;
    }

    int* bb = bbox + ((size_t)b * FP + f) * 4;
    bb[0]=pxmin; bb[1]=pxmax; bb[2]=pymin; bb[3]=pymax;
}

// ---------------------------------------------- async chunk stage (mem->LDS)
// Each thread DMAs one face record: 64B of plane coeffs + 16B bbox, using
// GLOBAL_LOAD_ASYNC_TO_LDS_B128 (GVS mode: saddr base + 32-bit vgpr offset;
// inst offset applies to both the LDS and global side). Tracked by ASYNCcnt.
__device__ __forceinline__ void rast_stage_async(
    const float* Pg, const int* Bg, int cbase, float* sPbuf, int* sBbuf)
{
    unsigned t     = threadIdx.x;
    unsigned ldsP  = (unsigned)(size_t)sPbuf + t * 64u;   // LDS byte addr
    unsigned ldsB  = (unsigned)(size_t)sBbuf + t * 16u;
    unsigned voffP = t * 64u;                             // global byte offset
    unsigned voffB = t * 16u;
    const void* gP = (const void*)(Pg + (size_t)cbase * 16);
    const void* gB = (const void*)(Bg + (size_t)cbase * 4);
    asm volatile(
        "global_load_async_to_lds_b128 %0, %1, %2 offset:0\n\t"
        "global_load_async_to_lds_b128 %0, %1, %2 offset:16\n\t"
        "global_load_async_to_lds_b128 %0, %1, %2 offset:32\n\t"
        "global_load_async_to_lds_b128 %0, %1, %2 offset:48"
        :: "v"(ldsP), "v"(voffP), "s"(gP) : "memory");
    asm volatile(
        "global_load_async_to_lds_b128 %0, %1, %2 offset:0"
        :: "v"(ldsB), "v"(voffB), "s"(gB) : "memory");
}

__device__ __forceinline__ void rast_wait_async0() {
    asm volatile("s_wait_asynccnt 0x0" ::: "memory");
}

// ----------------------------------------------------------- rasterizer
__global__ __launch_bounds__(256) void rast_main(
    const float* __restrict__ planes, const float* __restrict__ colw,
    const int* __restrict__ bbox, const int* __restrict__ gbox,
    float* __restrict__ out, int FP, int B)
{
    __shared__ __align__(16) float sP[2][FCHUNK * 16];
    __shared__ __align__(16) int   sB[2][FCHUNK * 4];

    const int lane = threadIdx.x & 31;
    const int wave = threadIdx.x >> 5;
    const int runsPerRow = WW / 16;            // 10
    const int runsPerBatch = HH * runsPerRow;  // 1600 (divisible by 8)

    int run = blockIdx.x * 8 + wave;
    int b  = run / runsPerBatch;
    int rr = run - b * runsPerBatch;
    int y  = rr / runsPerRow;
    int x0 = (rr - y * runsPerRow) * 16;

    // pix_ok (reference global-bbox gate), wave-uniform for an aligned 16-run
    int gxmin = gbox[b*4+0], gxmax = gbox[b*4+1];
    int gymin = gbox[b*4+2], gymax = gbox[b*4+3];
    int rx0 = max(gxmin - gxmin % BSZ, 0);
    int rx1 = min(gxmax, WW);
    int ry0 = max(gymin - gymin % BSZ, 0);
    int ry1 = min(gymax, HH);
    int bx = (x0 / BSZ) * BSZ;
    int by = (y  / BSZ) * BSZ;
    bool pix_ok = (bx >= rx0) && (bx < rx1) && (by >= ry0) && (by < ry1);

    // A fragment: lanes 0-15 = (X_m, Y), lanes 16-31 = (1, 0)   [16x4 f32 layout]
    v2f afrag;
    if (lane < 16) { afrag.x = (float)(x0 + lane); afrag.y = (float)y; }
    else           { afrag.x = 1.0f;               afrag.y = 0.0f;     }

    float bd[8]; int bi[8];
    #pragma unroll
    for (int r = 0; r < 8; ++r) { bd[r] = INF_V; bi[r] = 0; }

    const float* Pg = planes + (size_t)b * FP * 16;
    const int*   Bg = bbox   + (size_t)b * FP * 4;
    const int nsub = lane & 15;   // face column within 16-group
    const int hf   = lane >> 4;   // 0: K=0,1 coeffs / pixels 0-7 ; 1: K=2,3 / pixels 8-15

    // prologue: stage chunk 0 into buffer 0
    rast_stage_async(Pg, Bg, 0, sP[0], sB[0]);
    rast_wait_async0();
    __syncthreads();

    int cur = 0;
    for (int cbase = 0; cbase < FP; cbase += FCHUNK) {
        // kick off next chunk's DMA into the other buffer while we compute
        if (cbase + FCHUNK < FP)
            rast_stage_async(Pg, Bg, cbase + FCHUNK, sP[cur ^ 1], sB[cur ^ 1]);

        if (pix_ok) {
            const float* sPc = sP[cur];
            const int*   sBc = sB[cur];
            for (int g = 0; g < FCHUNK / 16; ++g) {
                int fl = g * 16 + nsub;
                const float* fp = sPc + fl * 16;
                int off = hf * 2;               // half0: (cx,cy) ; half1: (cc,0)
                v2f b0 = *(const v2f*)(fp + 0  + off);
                v2f b1 = *(const v2f*)(fp + 4  + off);
                v2f b2 = *(const v2f*)(fp + 8  + off);
                v2f b3 = *(const v2f*)(fp + 12 + off);

                v8f z = {0.f,0.f,0.f,0.f,0.f,0.f,0.f,0.f};
                v8f L0 = __builtin_amdgcn_wmma_f32_16x16x4_f32(false, afrag, false, b0, (short)0, z, false, false);
                v8f L1 = __builtin_amdgcn_wmma_f32_16x16x4_f32(false, afrag, false, b1, (short)0, z, false, false);
                v8f L2 = __builtin_amdgcn_wmma_f32_16x16x4_f32(false, afrag, false, b2, (short)0, z, false, false);
                v8f Dd = __builtin_amdgcn_wmma_f32_16x16x4_f32(false, afrag, false, b3, (short)0, z, false, false);

                v4i bb = *(const v4i*)(sBc + fl * 4);
                bool covB = (bb.y >= bx) && (bb.x < bx + BSZ) &&
                            (bb.w >= by) && (bb.z < by + BSZ);
                int fid = cbase + fl;

                #pragma unroll
                for (int r = 0; r < 8; ++r) {
                    bool cov = covB & (L0[r] >= 0.0f) & (L1[r] >= 0.0f) & (L2[r] >= 0.0f);
                    float D = Dd[r] + (cov ? 0.0f : INF_V);
                    D = (D == D) ? D : INF_V;                  // NaN -> INF_V
                    if (D < bd[r]) { bd[r] = D; bi[r] = fid; } // ties keep lowest fid
                }
            }
        }

        rast_wait_async0();   // our async issues for the next buffer are done
        __syncthreads();      // everyone's are visible; safe to swap
        cur ^= 1;
    }

    // cross-lane reduce: 16 faces per half-wave hold candidates for same pixels
    #pragma unroll
    for (int m = 1; m <= 8; m <<= 1) {
        #pragma unroll
        for (int r = 0; r < 8; ++r) {
            float od = __shfl_xor(bd[r], m, 32);
            int   oi = __shfl_xor(bi[r], m, 32);
            if (od < bd[r] || (od == bd[r] && oi < bi[r])) { bd[r] = od; bi[r] = oi; }
        }
    }

    // shade + write: lane 0 -> pixels 0-7, lane 16 -> pixels 8-15
    if (nsub == 0) {
        const float* Cg = colw + (size_t)b * FP * 12;
        size_t imgBase  = (size_t)b * 3 * HH * WW;
        size_t maskBase = (size_t)B * 3 * HH * WW + ((size_t)b * HH + y) * WW;
        #pragma unroll
        for (int r = 0; r < 8; ++r) {
            int px = x0 + hf * 8 + r;
            float X = (float)px, Y = (float)y;
            bool vis = bd[r] < LOW_INF;
            int idx = vis ? bi[r] : 0;
            const float* cc = Cg + (size_t)idx * 12;
            float o0 = vis ? (cc[0]*X + cc[1]*Y + cc[2])  : 0.0f;
            float o1 = vis ? (cc[4]*X + cc[5]*Y + cc[6])  : 0.0f;
            float o2 = vis ? (cc[8]*X + cc[9]*Y + cc[10]) : 0.0f;
            size_t pix = (size_t)y * WW + px;
            out[imgBase + 0*HH*WW + pix] = o0;
            out[imgBase + 1*HH*WW + pix] = o1;
            out[imgBase + 2*HH*WW + pix] = o2;
            out[maskBase + px] = vis ? 1.0f : 0.0f;
        }
    }
}

// ------------------------------------------------------------------ host
extern "C" void kernel_launch(void* const* d_in, const int* in_sizes, int n_in,
                              void* d_out, int out_size, void* d_ws, size_t ws_size,
                              hipStream_t stream) {
    const float* pt2d   = (const float*)d_in[0];
    const float* color  = (const float*)d_in[1];
    const float* pt3d   = (const float*)d_in[2];
    const float* normal = (const float*)d_in[3];
    const float* Rm     = (const float*)d_in[4];
    const float* Tm     = (const float*)d_in[5];
    const int*   face   = (const int*)d_in[6];

    int B    = in_sizes[4] / 9;                       // R is (B,3,3)
    int Fnum = in_sizes[6] / 3;                       // face is (3,F)
    int Vnum = in_sizes[0] / (3 * B);                 // pt_2d is (B,3,V)
    int FP   = (Fnum + FCHUNK - 1) / FCHUNK * FCHUNK; // padded face count

    float* planes = (float*)d_ws;                     // B*FP*16 f32
    float* colw   = planes + (size_t)B * FP * 16;     // B*FP*12 f32
    int*   bbox   = (int*)(colw + (size_t)B * FP * 12); // B*FP*4 i32
    int*   gbox   = bbox + (size_t)B * FP * 4;        // B*4 i32

    rast_init_gbox<<<1, 64, 0, stream>>>(gbox, B);

    rast_setup<<<(B * FP) / 256, 256, 0, stream>>>(
        pt2d, color, pt3d, normal, Rm, Tm, face,
        planes, colw, bbox, gbox, Fnum, FP, Vnum);

    int totalRuns = B * HH * (WW / 16);               // 3200, 8 waves/block
    rast_main<<<totalRuns / 8, 256, 0, stream>>>(
        planes, colw, bbox, gbox, (float*)d_out, FP, B);
}